// GCN_15418932593120
// MI455X (gfx1250) — compile-verified
//
#include <hip/hip_runtime.h>
#include <hip/hip_bf16.h>

// Problem constants (from reference)
#define NN 100000
#define EE 3200000
#define FF 128
#define HH 32
#define GG 512

typedef __attribute__((ext_vector_type(2))) float v2f;
typedef __attribute__((ext_vector_type(8))) float v8f;

// ---------------------------------------------------------------------------
// Kernel 1: degree count over real edges (self-loop folded in as +1 later)
// ---------------------------------------------------------------------------
__global__ void gcn_degree_kernel(const int* __restrict__ edge_index,
                                  float* __restrict__ deg) {
    int i = blockIdx.x * blockDim.x + threadIdx.x;
    if (i < EE) {
        int dst = edge_index[EE + i];   // row 1 of edge_index
        atomicAdd(&deg[dst], 1.0f);
    }
}

// ---------------------------------------------------------------------------
// Kernel 2: dinv = rsqrt(deg + 1)   (self-loop makes deg+1 >= 1 always)
// ---------------------------------------------------------------------------
__global__ void gcn_dinv_kernel(float* __restrict__ deg) {
    int i = blockIdx.x * blockDim.x + threadIdx.x;
    if (i < NN) deg[i] = rsqrtf(deg[i] + 1.0f);
}

// ---------------------------------------------------------------------------
// Kernel 3: h = x @ W1 via V_WMMA_F32_16X16X4_F32.
// One wave -> 16x32 output tile (two 16x16 accumulators).
// W1 staged transposed in LDS: Wt[n*128 + k] = W1[k*32 + n], so a lane's two
// K-consecutive B elements are a single 8B LDS load.
// ---------------------------------------------------------------------------
#define GEMM_WAVES 8
__global__ void gcn_gemm_wmma_kernel(const float* __restrict__ x,
                                     const float* __restrict__ W1,
                                     float* __restrict__ h) {
    __shared__ float Wt[HH * FF];   // 16 KB
    for (int i = threadIdx.x; i < HH * FF; i += blockDim.x) {
        int k = i >> 5;          // i = k*32 + n (W1 row-major [128,32])
        int n = i & 31;
        Wt[n * FF + k] = W1[i];
    }
    __syncthreads();

    int wave = threadIdx.x >> 5;
    int lane = threadIdx.x & 31;
    int rowTile = blockIdx.x * GEMM_WAVES + wave;   // 6250 tiles total (N/16 exact)
    if (rowTile >= NN / 16) return;                 // wave-uniform guard: EXEC stays all-1s

    int m  = lane & 15;          // A-matrix row within tile
    int kk = (lane >> 4) * 2;    // lanes 0-15 hold K=k,k+1 ; lanes 16-31 hold K=k+2,k+3
    int row0 = rowTile * 16;

    const float* xrow = x  + (size_t)(row0 + m) * FF;
    const float* b0p  = Wt + (size_t)m * FF;          // B tile n = 0..15
    const float* b1p  = Wt + (size_t)(m + 16) * FF;   // B tile n = 16..31

    v8f acc0 = {0.f,0.f,0.f,0.f,0.f,0.f,0.f,0.f};
    v8f acc1 = {0.f,0.f,0.f,0.f,0.f,0.f,0.f,0.f};

    #pragma unroll 4
    for (int k = 0; k < FF; k += 4) {
        v2f a  = *(const v2f*)(xrow + k + kk);
        v2f b0 = *(const v2f*)(b0p  + k + kk);
        v2f b1 = *(const v2f*)(b1p  + k + kk);
        acc0 = __builtin_amdgcn_wmma_f32_16x16x4_f32(false, a, false, b0,
                                                     (short)0, acc0, false, false);
        acc1 = __builtin_amdgcn_wmma_f32_16x16x4_f32(false, a, false, b1,
                                                     (short)0, acc1, false, false);
    }

    // C/D layout: VGPR r, lanes 0-15 -> M=r, N=lane ; lanes 16-31 -> M=r+8, N=lane-16
    int cn  = lane & 15;
    int cm0 = (lane >> 4) * 8;
    float* hb = h + (size_t)row0 * HH;
    #pragma unroll
    for (int r = 0; r < 8; ++r) {
        hb[(size_t)(cm0 + r) * HH + cn]      = acc0[r];
        hb[(size_t)(cm0 + r) * HH + cn + 16] = acc1[r];
    }
}

// ---------------------------------------------------------------------------
// Kernel 4: edge scatter. One wave32 per edge/self-loop, one lane per feature
// (H == 32 == wave width). acc[dst] += h[src] * dinv[src]*dinv[dst].
// ---------------------------------------------------------------------------
#define SCAT_WAVES 8
__global__ void gcn_scatter_kernel(const float* __restrict__ h,
                                   const int* __restrict__ edge_index,
                                   const float* __restrict__ dinv,
                                   float* __restrict__ acc) {
    long long job = (long long)blockIdx.x * SCAT_WAVES + (threadIdx.x >> 5);
    int lane = threadIdx.x & 31;
    if (job >= (long long)EE + NN) return;

    int src, dst;
    if (job < EE) {
        src = edge_index[job];          // row 0
        dst = edge_index[EE + job];     // row 1
    } else {
        src = dst = (int)(job - EE);    // self-loop
    }
    float w = dinv[src] * dinv[dst];
    float v = h[(size_t)src * HH + lane] * w;
    atomicAdd(&acc[(size_t)dst * HH + lane], v);
}

// ---------------------------------------------------------------------------
// Kernel 5: bias + ReLU + graph pooling (sum / count / max). One wave per node.
// Max via unsigned bit-pattern atomicMax (valid: post-ReLU values >= 0; zero
// init == reference's where(cnt>0, mx, 0)).
// ---------------------------------------------------------------------------
__global__ void gcn_pool_kernel(const float* __restrict__ acc,
                                const float* __restrict__ b1,
                                const int* __restrict__ batch,
                                float* __restrict__ s,
                                float* __restrict__ cnt,
                                unsigned int* __restrict__ mx) {
    long long node = (long long)blockIdx.x * SCAT_WAVES + (threadIdx.x >> 5);
    int lane = threadIdx.x & 31;
    if (node >= NN) return;

    int b = batch[node];
    float v = acc[(size_t)node * HH + lane] + b1[lane];
    v = fmaxf(v, 0.0f);
    atomicAdd(&s[(size_t)b * HH + lane], v);
    atomicMax(&mx[(size_t)b * HH + lane], __float_as_uint(v));
    if (lane == 0) atomicAdd(&cnt[b], 1.0f);
}

// ---------------------------------------------------------------------------
// Kernel 6: readout  out[g] = concat(s, s/cnt, mx) @ Wg + bg   (G=512, tiny)
// ---------------------------------------------------------------------------
__global__ void gcn_readout_kernel(const float* __restrict__ s,
                                   const float* __restrict__ cnt,
                                   const unsigned int* __restrict__ mx,
                                   const float* __restrict__ Wg,
                                   const float* __restrict__ bg,
                                   float* __restrict__ out) {
    int g = blockIdx.x * blockDim.x + threadIdx.x;
    if (g >= GG) return;
    float ic = 1.0f / fmaxf(cnt[g], 1.0f);
    float r = bg[0];
    #pragma unroll
    for (int f = 0; f < HH; ++f) {
        float sv = s[(size_t)g * HH + f];
        float mv = __uint_as_float(mx[(size_t)g * HH + f]);
        r += sv * Wg[f] + (sv * ic) * Wg[HH + f] + mv * Wg[2 * HH + f];
    }
    out[g] = r;
}

// ---------------------------------------------------------------------------
extern "C" void kernel_launch(void* const* d_in, const int* in_sizes, int n_in,
                              void* d_out, int out_size, void* d_ws, size_t ws_size,
                              hipStream_t stream) {
    const float* x          = (const float*)d_in[0];   // [N,128]
    const int*   edge_index = (const int*)  d_in[1];   // [2,E]
    const int*   batch      = (const int*)  d_in[2];   // [N]
    const float* W1         = (const float*)d_in[3];   // [128,32]
    const float* b1         = (const float*)d_in[4];   // [32]
    const float* Wg         = (const float*)d_in[5];   // [96,1]
    const float* bg         = (const float*)d_in[6];   // [1]
    float* out = (float*)d_out;                        // [512]

    // Workspace layout
    float* deg = (float*)d_ws;                  // N        (becomes dinv in place)
    float* h   = deg + NN;                      // N*32
    float* acc = h + (size_t)NN * HH;           // N*32
    float* s   = acc + (size_t)NN * HH;         // G*32
    float* cnt = s + (size_t)GG * HH;           // G
    unsigned int* mx = (unsigned int*)(cnt + GG); // G*32

    // Zero accumulators every call (deterministic across graph replays)
    hipMemsetAsync(deg, 0, sizeof(float) * NN, stream);
    hipMemsetAsync(acc, 0, sizeof(float) * (size_t)NN * HH, stream);
    hipMemsetAsync(s,   0, sizeof(float) * ((size_t)GG * HH + GG + (size_t)GG * HH), stream);

    // 1) degrees
    gcn_degree_kernel<<<(EE + 255) / 256, 256, 0, stream>>>(edge_index, deg);
    // 2) dinv = rsqrt(deg+1)
    gcn_dinv_kernel<<<(NN + 255) / 256, 256, 0, stream>>>(deg);
    // 3) h = x @ W1 (WMMA fp32)
    {
        int tiles = NN / 16;                            // 6250, exact
        int blocks = (tiles + GEMM_WAVES - 1) / GEMM_WAVES;
        gcn_gemm_wmma_kernel<<<blocks, GEMM_WAVES * 32, 0, stream>>>(x, W1, h);
    }
    // 4) edge + self-loop scatter (one wave per edge)
    {
        long long jobs = (long long)EE + NN;
        int blocks = (int)((jobs + SCAT_WAVES - 1) / SCAT_WAVES);
        gcn_scatter_kernel<<<blocks, SCAT_WAVES * 32, 0, stream>>>(h, edge_index, deg, acc);
    }
    // 5) bias + relu + pooling (one wave per node)
    {
        int blocks = (NN + SCAT_WAVES - 1) / SCAT_WAVES;
        gcn_pool_kernel<<<blocks, SCAT_WAVES * 32, 0, stream>>>(acc, b1, batch, s, cnt, mx);
    }
    // 6) readout
    gcn_readout_kernel<<<(GG + 255) / 256, 256, 0, stream>>>(s, cnt, mx, Wg, bg, out);
}